// CLIR_3178275799225
// MI455X (gfx1250) — compile-verified
//
#include <hip/hip_runtime.h>

#define N_ROWS 2000000
#define P_DIM 64
#define C_CLS 100
#define ACC_N (C_CLS * P_DIM)   // 6400 per quantity

typedef __attribute__((ext_vector_type(2))) float v2f;
typedef __attribute__((ext_vector_type(8))) float v8f;

// ---------------- kernel 0: zero workspace accumulators ----------------
__global__ void vc_zero_ws(float* __restrict__ ws, int n) {
  int i = blockIdx.x * blockDim.x + threadIdx.x;
  if (i < n) ws[i] = 0.0f;
}

// ---------------- kernel 1: streaming segmented accumulation ----------------
// ws layout (floats): [0,6400)=s  [6400,12800)=ss  [12800,12900)=counts
__global__ __launch_bounds__(256) void vc_accum(const float* __restrict__ x,
                                                const int* __restrict__ t,
                                                float* __restrict__ ws) {
  __shared__ float s_sh[ACC_N];
  __shared__ float ss_sh[ACC_N];
  __shared__ float cnt_sh[C_CLS];

  const int tid = threadIdx.x;
  for (int i = tid; i < ACC_N; i += 256) { s_sh[i] = 0.0f; ss_sh[i] = 0.0f; }
  if (tid < C_CLS) cnt_sh[tid] = 0.0f;
  __syncthreads();

  const int lane  = tid & 31;
  const int wave  = tid >> 5;
  const int gwave = blockIdx.x * 8 + wave;          // 8 wave32 per block
  const int nwave = gridDim.x * 8;
  const int strideRows = nwave * 32;                // rows per grid pass

  for (int base = gwave * 32; base < N_ROWS; base += strideRows) {
    // one label per lane (N divisible by 32, so full blocks)
    unsigned uc = (unsigned)t[base + lane];
    if (uc >= (unsigned)C_CLS) uc = 0u;             // clamp: never index OOB
    atomicAdd(&cnt_sh[uc], 1.0f);                   // 1 ds op per 32 rows/lane

    // prefetch next row block for this wave (32 rows * 256B = 8KB)
    int nb = base + strideRows;
    if (nb < N_ROWS) {
      const float* np = x + (size_t)nb * P_DIM;
      __builtin_prefetch(np + lane * 64, 0, 1);     // lane*256B -> covers 8KB
    }

#pragma unroll
    for (int k = 0; k < 32; ++k) {
      // broadcast row k's class via v_readlane (keeps DS pipe free)
      int c = __builtin_amdgcn_readlane((int)uc, k);
      const float* rowp = x + (size_t)(base + k) * P_DIM;
      float a = rowp[lane];          // coalesced 128B
      float b = rowp[lane + 32];     // coalesced 128B
      int o = c * P_DIM + lane;      // lane->bank: conflict-free ds_add_f32
      atomicAdd(&s_sh[o],       a);
      atomicAdd(&s_sh[o + 32],  b);
      atomicAdd(&ss_sh[o],      a * a);
      atomicAdd(&ss_sh[o + 32], b * b);
    }
  }
  __syncthreads();

  float* gs   = ws;
  float* gss  = ws + ACC_N;
  float* gcnt = ws + 2 * ACC_N;
  for (int i = tid; i < ACC_N; i += 256) {
    atomicAdd(&gs[i],  s_sh[i]);
    atomicAdd(&gss[i], ss_sh[i]);
  }
  if (tid < C_CLS) atomicAdd(&gcnt[tid], cnt_sh[tid]);
}

// ---------------- kernel 2: variance + WMMA ones-matmul reduction ----------------
__global__ __launch_bounds__(256) void vc_finalize(const float* __restrict__ ws,
                                                   float* __restrict__ out) {
  __shared__ float var_sh[ACC_N];
  __shared__ float red[32];

  const float* gs   = ws;
  const float* gss  = ws + ACC_N;
  const float* gcnt = ws + 2 * ACC_N;
  const int tid = threadIdx.x;

  for (int i = tid; i < ACC_N; i += 256) {
    int   c = i >> 6;                   // i / 64
    float n = gcnt[c];
    float s = gs[i];
    float q = gss[i];
    var_sh[i] = (q - s * s / n) / (n - 1.0f);
  }
  __syncthreads();

  if (tid < 32) {                       // wave 0: EXEC all-ones -> WMMA legal
    // A-matrix 16x4 f32 layout: lanes 0-15 hold K=0,1; lanes 16-31 hold K=2,3
    const int m  = tid & 15;
    const int k0 = (tid >> 4) * 2;
    v8f acc = {0.f, 0.f, 0.f, 0.f, 0.f, 0.f, 0.f, 0.f};
    v2f ones; ones.x = 1.0f; ones.y = 1.0f;   // B = 4x16 all-ones (layout-invariant)
    for (int tile = 0; tile < C_CLS; ++tile) {
      v2f am;
      int bi = tile * 64 + m * 4 + k0;        // A[M][K] = var_sh[tile*64 + M*4 + K]
      am.x = var_sh[bi];
      am.y = var_sh[bi + 1];
      // D[m][n] += sum_k A[m][k] : row-sum reduction on the matrix pipe
      acc = __builtin_amdgcn_wmma_f32_16x16x4_f32(
          /*neg_a=*/false, am, /*neg_b=*/false, ones,
          /*c_mod=*/(short)0, acc, /*reuse_a=*/false, /*reuse_b=*/false);
    }
    // column n = (lane%16) of D: lane 0 holds M=0..7, lane 16 holds M=8..15
    red[tid] = acc[0] + acc[1] + acc[2] + acc[3] + acc[4] + acc[5] + acc[6] + acc[7];
  }
  __syncthreads();
  if (tid == 0) out[0] = (red[0] + red[16]) / (float)C_CLS;
}

// ---------------- launch ----------------
extern "C" void kernel_launch(void* const* d_in, const int* in_sizes, int n_in,
                              void* d_out, int out_size, void* d_ws, size_t ws_size,
                              hipStream_t stream) {
  const float* x = (const float*)d_in[0];
  const int*   t = (const int*)d_in[1];
  float* ws  = (float*)d_ws;
  float* out = (float*)d_out;

  const int wsN = 2 * ACC_N + C_CLS;   // 12900 floats (~51.6 KB)
  vc_zero_ws<<<(wsN + 255) / 256, 256, 0, stream>>>(ws, wsN);
  vc_accum<<<512, 256, 0, stream>>>(x, t, ws);
  vc_finalize<<<1, 256, 0, stream>>>(ws, out);
}